// STRC_16604343566780
// MI455X (gfx1250) — compile-verified
//
#include <hip/hip_runtime.h>
#include <math.h>

#define D 128

typedef __attribute__((ext_vector_type(2))) float v2f;
typedef __attribute__((ext_vector_type(8))) float v8f;

// ---------------------------------------------------------------------------
// SpMM: one wave32 per edge. Lane l handles channels [4l, 4l+4).
// T[row, :] += val * X[col, :]   via global_atomic_add_f32.
// X (51 MB) and T (51 MB) both fit in the 192 MB L2, so the random gather and
// the scatter-atomics are L2 traffic, not HBM.
// ---------------------------------------------------------------------------
__global__ void spmm_edges(const float* __restrict__ X,
                           const float* __restrict__ evals,
                           const int*   __restrict__ eidx,
                           float*       __restrict__ T,
                           int E) {
    int gtid = blockIdx.x * blockDim.x + threadIdx.x;
    int e    = gtid >> 5;
    int lane = threadIdx.x & 31;
    if (e >= E) return;

    // stream-ahead prefetch of the edge arrays (emits global_prefetch_b8)
    if (e + 2048 < E) {
        __builtin_prefetch(&evals[e + 2048], 0, 1);
        __builtin_prefetch(&eidx[e + 2048], 0, 1);
        __builtin_prefetch(&eidx[E + e + 2048], 0, 1);
    }

    float val = evals[e];
    int   row = eidx[e];
    int   col = eidx[E + e];

    float4 x = *((const float4*)(X + (size_t)col * D) + lane);

    float* t = T + (size_t)row * D + lane * 4;
    atomicAdd(t + 0, val * x.x);
    atomicAdd(t + 1, val * x.y);
    atomicAdd(t + 2, val * x.z);
    atomicAdd(t + 3, val * x.w);
}

// ---------------------------------------------------------------------------
// Per-channel sum and sum-of-squares of T[N, 128] using V_WMMA_F32_16X16X4_F32:
//   D = ones(16x4) @ B(4x16) + C  -> every output row = column sums of B tile.
// One wave per (16-column tile, row-slice); accumulate N/(4*gridDim.y) tiles per wave in
// the f32 C operand, then drain with 16 atomics.
// B-operand ISA layout (4x16 f32): VGPR0 = K=0 (lanes 0-15) / K=2 (lanes 16-31),
//                                  VGPR1 = K=1 (lanes 0-15) / K=3 (lanes 16-31).
// blockDim = 32 (all lanes active: WMMA requires EXEC all-ones).
// ---------------------------------------------------------------------------
__global__ void colstats_wmma(const float* __restrict__ T, int N,
                              float* __restrict__ S, float* __restrict__ Q) {
    int lane    = threadIdx.x;            // 0..31
    int colBase = blockIdx.x * 16;        // 8 column tiles for D=128
    int half    = lane >> 4;              // selects K-pair {0,1} or {2,3}
    int c       = lane & 15;

    const float* base = T + colBase + c;

    v2f ones; ones[0] = 1.0f; ones[1] = 1.0f;
    v8f acc  = {};   // column sums
    v8f acc2 = {};   // column sums of squares

    int step = gridDim.y * 4;
    for (int n = blockIdx.y * 4; n + 3 < N; n += step) {
        const float* p = base + (size_t)(n + 2 * half) * D;
        v2f b, b2;
        b[0]  = p[0];          // row n + 2*half     (K = 2*half)
        b[1]  = p[D];          // row n + 2*half + 1 (K = 2*half + 1)
        b2[0] = b[0] * b[0];
        b2[1] = b[1] * b[1];
        acc  = __builtin_amdgcn_wmma_f32_16x16x4_f32(false, ones, false, b,
                                                     (short)0, acc,  false, false);
        acc2 = __builtin_amdgcn_wmma_f32_16x16x4_f32(false, ones, false, b2,
                                                     (short)0, acc2, false, false);
    }

    // C/D layout: VGPR0 lanes 0-15 hold row M=0, N = lane. All rows identical.
    if (lane < 16) {
        atomicAdd(&S[colBase + lane], acc[0]);
        atomicAdd(&Q[colBase + lane], acc2[0]);
    }
}

// ---------------------------------------------------------------------------
// Fold BN (train-mode batch stats, biased var, eps=1e-5) into per-channel
// scale/shift:  y = x*scale + shift.
// ---------------------------------------------------------------------------
__global__ void bn_finalize(const float* __restrict__ S, const float* __restrict__ Q,
                            const float* __restrict__ gamma, const float* __restrict__ beta,
                            float* __restrict__ scale, float* __restrict__ shift, int N) {
    int d = threadIdx.x;           // 128 threads
    float invN = 1.0f / (float)N;
    float mu   = S[d] * invN;
    float var  = Q[d] * invN - mu * mu;
    float g    = gamma[d] * rsqrtf(var + 1e-5f);
    scale[d] = g;
    shift[d] = beta[d] - mu * g;
}

// ---------------------------------------------------------------------------
// Apply BN elementwise; optionally stash normalized X (needed as next power's
// input) and accumulate 0.5 * X into the final output.
// ---------------------------------------------------------------------------
__global__ void bn_apply(const float4* __restrict__ T4,
                         const float*  __restrict__ scale,
                         const float*  __restrict__ shift,
                         float4* __restrict__ Xout,   // may be null (last power)
                         float4* __restrict__ out,
                         int n4, int accum) {
    int gi = blockIdx.x * blockDim.x + threadIdx.x;
    if (gi >= n4) return;
    int q = gi & (D / 4 - 1);

    float4 t  = T4[gi];
    float4 sc = ((const float4*)scale)[q];
    float4 sh = ((const float4*)shift)[q];

    float4 y;
    y.x = fmaf(t.x, sc.x, sh.x);
    y.y = fmaf(t.y, sc.y, sh.y);
    y.z = fmaf(t.z, sc.z, sh.z);
    y.w = fmaf(t.w, sc.w, sh.w);

    if (Xout) Xout[gi] = y;

    float4 h = make_float4(0.5f * y.x, 0.5f * y.y, 0.5f * y.z, 0.5f * y.w);
    if (accum) {
        float4 o = out[gi];
        o.x += h.x; o.y += h.y; o.z += h.z; o.w += h.w;
        out[gi] = o;
    } else {
        out[gi] = h;
    }
}

extern "C" void kernel_launch(void* const* d_in, const int* in_sizes, int n_in,
                              void* d_out, int out_size, void* d_ws, size_t ws_size,
                              hipStream_t stream) {
    const float* evals = (const float*)d_in[0];   // [E]
    const float* W     = (const float*)d_in[1];   // [N, 128]
    const float* bnw   = (const float*)d_in[2];   // [2, 128]
    const float* bnb   = (const float*)d_in[3];   // [2, 128]
    const int*   eidx  = (const int*)  d_in[4];   // [2, E] (row; col)

    int E = in_sizes[0];
    int N = in_sizes[1] / D;

    float* out = (float*)d_out;

    // Workspace layout: T (N*D f32) | X1 (N*D f32) | S,Q,scale,shift (4*128 f32)
    char*  ws     = (char*)d_ws;
    size_t tBytes = (size_t)N * D * sizeof(float);
    float* T     = (float*)ws;
    float* X1    = (float*)(ws + tBytes);
    float* S     = (float*)(ws + 2 * tBytes);
    float* Q     = S + D;
    float* scale = S + 2 * D;
    float* shift = S + 3 * D;

    int spmmBlocks  = (int)(((size_t)E * 32 + 255) / 256);
    int n4          = N * (D / 4);
    int applyBlocks = (n4 + 255) / 256;

    for (int p = 0; p < 2; ++p) {
        const float* Xin = (p == 0) ? W : X1;
        hipMemsetAsync(T, 0, tBytes, stream);
        hipMemsetAsync(S, 0, 2 * D * sizeof(float), stream);

        spmm_edges<<<spmmBlocks, 256, 0, stream>>>(Xin, evals, eidx, T, E);

        colstats_wmma<<<dim3(D / 16, 256), 32, 0, stream>>>(T, N, S, Q);

        bn_finalize<<<1, D, 0, stream>>>(S, Q, bnw + p * D, bnb + p * D,
                                         scale, shift, N);

        bn_apply<<<applyBlocks, 256, 0, stream>>>(
            (const float4*)T, scale, shift,
            (p == 0) ? (float4*)X1 : (float4*)nullptr,
            (float4*)out, n4, p);
    }
}